// Gate3_80891414052934
// MI455X (gfx1250) — compile-verified
//
#include <hip/hip_runtime.h>

// Gate3: r[b,c] = gate(e1[b,c]) + gate(e2[b,c]) + gate(x3[b,c]), c = 0..15
//   e1 = repeat(x1, 4, axis=1), e2 = repeat(x2, 2, axis=1)
//   gate(x) = sigmoid(x*w + b0) * x
// Memory-bound streaming kernel: 112 B in + 64 B out per row, zero reuse.
// -> 128-bit non-temporal loads/stores, 28 sigmoids/row (broadcast reuse),
//    fast sigmoid via v_exp_f32 (2^x) + v_rcp_f32.

typedef float f4 __attribute__((ext_vector_type(4)));

__device__ __forceinline__ float gate_f(float x, float w, float b0) {
    // sigmoid(t) = 1 / (1 + 2^(-t*log2(e)))
    float t = __builtin_fmaf(x, w, b0);
    float e = __builtin_amdgcn_exp2f(t * -1.4426950408889634f); // v_exp_f32
    float s = __builtin_amdgcn_rcpf(1.0f + e);                  // v_rcp_f32
    return s * x;
}

__global__ __launch_bounds__(256) void gate3_kernel(
    const float* __restrict__ x1,   // (rows, 4)
    const float* __restrict__ x2,   // (rows, 8)
    const float* __restrict__ x3,   // (rows, 16)
    const float* __restrict__ W,    // (1,1)
    const float* __restrict__ bia,  // (1,)
    float* __restrict__ out,        // (rows, 16)
    int rows)
{
    const int row = blockIdx.x * blockDim.x + threadIdx.x;
    if (row >= rows) return;

    // Uniform scalar params -> s_load path
    const float w  = W[0];
    const float b0 = bia[0];

    const f4* p1 = reinterpret_cast<const f4*>(x1) + row;       // 1 x b128
    const f4* p2 = reinterpret_cast<const f4*>(x2) + row * 2;   // 2 x b128
    const f4* p3 = reinterpret_cast<const f4*>(x3) + row * 4;   // 4 x b128
    f4*       po = reinterpret_cast<f4*>(out)      + row * 4;   // 4 x b128

    // Non-temporal: 369 MB stream, no reuse, don't pollute L2.
    f4 a  = __builtin_nontemporal_load(p1);
    f4 c0 = __builtin_nontemporal_load(p2 + 0);
    f4 c1 = __builtin_nontemporal_load(p2 + 1);
    f4 d0 = __builtin_nontemporal_load(p3 + 0);
    f4 d1 = __builtin_nontemporal_load(p3 + 1);
    f4 d2 = __builtin_nontemporal_load(p3 + 2);
    f4 d3 = __builtin_nontemporal_load(p3 + 3);

    // gate(x1): each value reused for 4 output columns
    float g1[4];
#pragma unroll
    for (int j = 0; j < 4; ++j) g1[j] = gate_f(a[j], w, b0);

    // gate(x2): each value reused for 2 output columns
    float g2[8];
#pragma unroll
    for (int j = 0; j < 4; ++j) g2[j]     = gate_f(c0[j], w, b0);
#pragma unroll
    for (int j = 0; j < 4; ++j) g2[4 + j] = gate_f(c1[j], w, b0);

    f4 r0, r1, r2, r3;
#pragma unroll
    for (int j = 0; j < 4; ++j) {
        const int c0i = j;          // cols 0..3
        const int c1i = 4 + j;      // cols 4..7
        const int c2i = 8 + j;      // cols 8..11
        const int c3i = 12 + j;     // cols 12..15
        r0[j] = g1[c0i >> 2] + g2[c0i >> 1] + gate_f(d0[j], w, b0);
        r1[j] = g1[c1i >> 2] + g2[c1i >> 1] + gate_f(d1[j], w, b0);
        r2[j] = g1[c2i >> 2] + g2[c2i >> 1] + gate_f(d2[j], w, b0);
        r3[j] = g1[c3i >> 2] + g2[c3i >> 1] + gate_f(d3[j], w, b0);
    }

    __builtin_nontemporal_store(r0, po + 0);
    __builtin_nontemporal_store(r1, po + 1);
    __builtin_nontemporal_store(r2, po + 2);
    __builtin_nontemporal_store(r3, po + 3);
}

extern "C" void kernel_launch(void* const* d_in, const int* in_sizes, int n_in,
                              void* d_out, int out_size, void* d_ws, size_t ws_size,
                              hipStream_t stream) {
    (void)n_in; (void)out_size; (void)d_ws; (void)ws_size;
    const float* x1  = (const float*)d_in[0];
    const float* x2  = (const float*)d_in[1];
    const float* x3  = (const float*)d_in[2];
    const float* W   = (const float*)d_in[3];
    const float* bia = (const float*)d_in[4];
    float* out = (float*)d_out;

    const int rows = in_sizes[0] / 4;           // x1 is (rows, 4)
    const int block = 256;                      // 8 wave32 per workgroup
    const int grid = (rows + block - 1) / block;

    gate3_kernel<<<grid, block, 0, stream>>>(x1, x2, x3, W, bia, out, rows);
}